// Mesh2SDF_1211180777899
// MI455X (gfx1250) — compile-verified
//
#include <hip/hip_runtime.h>
#include <math.h>

typedef float v2f __attribute__((ext_vector_type(2)));
typedef float v8f __attribute__((ext_vector_type(8)));

// Per-triangle 128-byte record (8 float4 slots, 6 used):
//  f0 = ( abx, aby, abz, -a.ab )        -> B operand for d1  = P.ab - a.ab
//  f1 = ( acx, acy, acz, -a.ac )        -> B operand for d2  = P.ac - a.ac
//  f2 = (-2ax,-2ay,-2az,  a.a  )        -> B operand for ap2 = |P|^2 - 2P.a + a.a
//  f3 = (  nx,  ny,  nz, -a.n  )        -> B operand for side = P.n - a.n
//  f4 = ( b.ab-a.ab, b.ac-a.ac, c.ab-a.ab, c.ac-a.ac )   (d3,d4,d5,d6 offsets)
//  f5 = ( |ab|^2, |ac|^2, ab.ac, 0 )
#define REC4 8   // float4 slots per triangle (128B aligned)

// ---------------------------------------------------------------------------
__global__ void mesh2sdf_tri_pre(const float* __restrict__ verts,
                                 const int*   __restrict__ faces,
                                 float4*      __restrict__ rec,
                                 int F) {
  int t = blockIdx.x * blockDim.x + threadIdx.x;
  if (t >= F) return;
  int i0 = faces[t * 3 + 0];
  int i1 = faces[t * 3 + 1];
  int i2 = faces[t * 3 + 2];
  float ax = verts[i0 * 3 + 0], ay = verts[i0 * 3 + 1], az = verts[i0 * 3 + 2];
  float bx = verts[i1 * 3 + 0], by = verts[i1 * 3 + 1], bz = verts[i1 * 3 + 2];
  float cx = verts[i2 * 3 + 0], cy = verts[i2 * 3 + 1], cz = verts[i2 * 3 + 2];

  float abx = bx - ax, aby = by - ay, abz = bz - az;
  float acx = cx - ax, acy = cy - ay, acz = cz - az;
  float nx = aby * acz - abz * acy;
  float ny = abz * acx - abx * acz;
  float nz = abx * acy - aby * acx;

  float aab = ax * abx + ay * aby + az * abz;
  float aac = ax * acx + ay * acy + az * acz;
  float bab = bx * abx + by * aby + bz * abz;
  float bac = bx * acx + by * acy + bz * acz;
  float cab = cx * abx + cy * aby + cz * abz;
  float cac = cx * acx + cy * acy + cz * acz;

  float4* r = rec + (size_t)t * REC4;
  r[0] = make_float4( abx,  aby,  abz, -aab);
  r[1] = make_float4( acx,  acy,  acz, -aac);
  r[2] = make_float4(-2.0f * ax, -2.0f * ay, -2.0f * az,
                     ax * ax + ay * ay + az * az);
  r[3] = make_float4( nx, ny, nz, -(ax * nx + ay * ny + az * nz));
  r[4] = make_float4( bab - aab, bac - aac, cab - aab, cac - aac);
  r[5] = make_float4( abx * abx + aby * aby + abz * abz,
                      acx * acx + acy * acy + acz * acz,
                      abx * acx + aby * acy + abz * acz, 0.0f);
}

// ---------------------------------------------------------------------------
// Each wave32 owns 32 grid points (2 WMMA row tiles); loops over 16-triangle
// tiles. Per tile: 6 clause-able b128 loads + 8 x v_wmma_f32_16x16x4_f32
// (A = [x y z 1] rows, B = folded triangle columns), then per-pair region
// classification on the accumulator elements.
// D layout: element v of lane l -> point (v + 8*(l>=16)), triangle (l&15).
// ---------------------------------------------------------------------------
__global__ __launch_bounds__(256)
void mesh2sdf_main(const float4* __restrict__ rec,
                   float*        __restrict__ out,
                   int F, int D, int M) {
  const int lane = threadIdx.x & 31;
  const int wave = threadIdx.x >> 5;
  const int half = lane >> 4;               // 0: K0/K1 (x,y); 1: K2/K3 (z,1)
  const int j    = lane & 15;               // A row / B column sub-index
  const int base = (blockIdx.x * 8 + wave) * 32;   // 32 points per wave

  const float inv = 2.0f / (float)(D - 1);
  const int   DD  = D * D;

  v2f   A[2];
  float p2loc[2][8];
#pragma unroll
  for (int pt = 0; pt < 2; ++pt) {
    int pm = base + pt * 16 + j; pm = pm < M ? pm : M - 1;
    int ix = pm / DD, iy = (pm / D) % D, iz = pm % D;
    float px = -1.0f + inv * (float)ix;
    float py = -1.0f + inv * (float)iy;
    float pz = -1.0f + inv * (float)iz;
    A[pt].x = half ? pz   : px;
    A[pt].y = half ? 1.0f : py;             // K3 = 1 multiplies folded consts
#pragma unroll
    for (int v = 0; v < 8; ++v) {
      int q = base + pt * 16 + v + 8 * half; q = q < M ? q : M - 1;
      int qx = q / DD, qy = (q / D) % D, qz = q % D;
      float fx = -1.0f + inv * (float)qx;
      float fy = -1.0f + inv * (float)qy;
      float fz = -1.0f + inv * (float)qz;
      p2loc[pt][v] = fx * fx + fy * fy + fz * fz;
    }
  }

  float bestD[2][8], bestS[2][8];
#pragma unroll
  for (int pt = 0; pt < 2; ++pt)
#pragma unroll
    for (int v = 0; v < 8; ++v) { bestD[pt][v] = 1.0e10f; bestS[pt][v] = 1.0f; }

  const float eps = 1e-12f;

  for (int t0 = 0; t0 < F; t0 += 16) {
    int t = t0 + j; t = t < F ? t : F - 1;
    const float4* r = rec + (size_t)t * REC4;

    if (t0 + 16 < F) {
      int tn = t0 + 16 + j; tn = tn < F ? tn : F - 1;
      __builtin_prefetch(rec + (size_t)tn * REC4, 0, 3);
    }

    float4 f0 = r[0], f1 = r[1], f2 = r[2], f3 = r[3], f4 = r[4], f5 = r[5];

    v2f Bab, Bac, Ba, Bn;
    Bab.x = half ? f0.z : f0.x;  Bab.y = half ? f0.w : f0.y;
    Bac.x = half ? f1.z : f1.x;  Bac.y = half ? f1.w : f1.y;
    Ba.x  = half ? f2.z : f2.x;  Ba.y  = half ? f2.w : f2.y;
    Bn.x  = half ? f3.z : f3.x;  Bn.y  = half ? f3.w : f3.y;

    v8f Cab[2], Cac[2], Ca[2], Cn[2];
#pragma unroll
    for (int pt = 0; pt < 2; ++pt) {
      v8f z = {0.f,0.f,0.f,0.f,0.f,0.f,0.f,0.f};
      Cab[pt] = __builtin_amdgcn_wmma_f32_16x16x4_f32(false, A[pt], false, Bab, (short)0, z, false, false);
      Cac[pt] = __builtin_amdgcn_wmma_f32_16x16x4_f32(false, A[pt], false, Bac, (short)0, z, false, false);
      Ca[pt]  = __builtin_amdgcn_wmma_f32_16x16x4_f32(false, A[pt], false, Ba,  (short)0, z, false, false);
      Cn[pt]  = __builtin_amdgcn_wmma_f32_16x16x4_f32(false, A[pt], false, Bn,  (short)0, z, false, false);
    }

    const float babp = f4.x, bacp = f4.y, cabp = f4.z, cacp = f4.w;
    const float abab = f5.x, acac = f5.y, abac = f5.z;

#pragma unroll
    for (int pt = 0; pt < 2; ++pt) {
#pragma unroll
      for (int v = 0; v < 8; ++v) {
        float d1 = Cab[pt][v];            // P.ab - a.ab (folded)
        float d2 = Cac[pt][v];            // P.ac - a.ac (folded)
        float d3 = d1 - babp;
        float d4 = d2 - bacp;
        float d5 = d1 - cabp;
        float d6 = d2 - cacp;

        float va = d3 * d6 - d5 * d4;
        float vb = d5 * d2 - d1 * d6;
        float vc = d1 * d4 - d3 * d2;

        float rden = __builtin_amdgcn_rcpf(fmaxf(va + vb + vc, eps));
        float vv = vb * rden;
        float wq = vc * rden;

        float e1 = d4 - d3, e2 = d5 - d6;
        float wbc = e1 * __builtin_amdgcn_rcpf(fmaxf(e1 + e2, eps));
        bool cBC = (va <= 0.f) && (e1 >= 0.f) && (e2 >= 0.f);
        vv = cBC ? 1.0f - wbc : vv;  wq = cBC ? wbc : wq;

        float wac = d2 * __builtin_amdgcn_rcpf(fmaxf(d2 - d6, eps));
        bool cAC = (vb <= 0.f) && (d2 >= 0.f) && (d6 <= 0.f);
        vv = cAC ? 0.0f : vv;        wq = cAC ? wac : wq;

        float vab = d1 * __builtin_amdgcn_rcpf(fmaxf(d1 - d3, eps));
        bool cAB = (vc <= 0.f) && (d1 >= 0.f) && (d3 <= 0.f);
        vv = cAB ? vab : vv;         wq = cAB ? 0.0f : wq;

        bool cC = (d6 >= 0.f) && (d5 <= d6);
        vv = cC ? 0.0f : vv;         wq = cC ? 1.0f : wq;

        bool cB = (d3 >= 0.f) && (d4 <= d3);
        vv = cB ? 1.0f : vv;         wq = cB ? 0.0f : wq;

        bool cA = (d1 <= 0.f) && (d2 <= 0.f);
        vv = cA ? 0.0f : vv;         wq = cA ? 0.0f : wq;

        float ap2   = p2loc[pt][v] + Ca[pt][v];   // |P|^2 - 2P.a + a.a
        float dist2 = ap2 - 2.0f * vv * d1 - 2.0f * wq * d2
                    + vv * vv * abab + 2.0f * vv * wq * abac + wq * wq * acac;
        dist2 = fmaxf(dist2, 0.0f);
        float side = Cn[pt][v];                   // P.n - a.n (folded)

        if (dist2 < bestD[pt][v]) { bestD[pt][v] = dist2; bestS[pt][v] = side; }
      }
    }
  }

  // Min-reduce over the 16 triangle lanes of each half (wave32 butterfly;
  // xor masks 1..8 never cross the 16-lane halves).
#pragma unroll
  for (int pt = 0; pt < 2; ++pt) {
#pragma unroll
    for (int v = 0; v < 8; ++v) {
      float bd = bestD[pt][v], bs = bestS[pt][v];
#pragma unroll
      for (int m2 = 1; m2 < 16; m2 <<= 1) {
        float od = __shfl_xor(bd, m2, 32);
        float os = __shfl_xor(bs, m2, 32);
        if (od < bd) { bd = od; bs = os; }
      }
      if (j == 0) {
        int q = base + pt * 16 + v + 8 * half;
        if (q < M) out[q] = (bs < 0.0f ? -1.0f : 1.0f) * sqrtf(bd);
      }
    }
  }
}

// ---------------------------------------------------------------------------
extern "C" void kernel_launch(void* const* d_in, const int* in_sizes, int n_in,
                              void* d_out, int out_size, void* d_ws, size_t ws_size,
                              hipStream_t stream) {
  (void)n_in; (void)ws_size;
  const float* verts = (const float*)d_in[0];
  const int*   faces = (const int*)d_in[1];

  int F = in_sizes[1] / 3;
  int M = out_size;
  int D = (int)lroundf(cbrtf((float)M));

  float4* rec = (float4*)d_ws;   // F * 128 bytes
  float*  out = (float*)d_out;

  mesh2sdf_tri_pre<<<(F + 255) / 256, 256, 0, stream>>>(verts, faces, rec, F);

  int nwaves = (M + 31) / 32;        // 32 points per wave
  int blocks = (nwaves + 7) / 8;     // 8 waves per block
  mesh2sdf_main<<<blocks, 256, 0, stream>>>(rec, out, F, D, M);
}